// DifferentiablePianoSynth_11854109737163
// MI455X (gfx1250) — compile-verified
//
#include <hip/hip_runtime.h>
#include <stdint.h>

#define TWO_PI 6.28318530717958647692f
#define INV_SR (1.0f / 48000.0f)
#define WIN 1024

typedef __attribute__((ext_vector_type(2))) float v2f;
typedef __attribute__((ext_vector_type(8))) float v8f;
typedef __attribute__((address_space(3))) float lds_f32;

// ---------------- JAX threefry2x32 (exact algorithm, 20 rounds) ----------------
__device__ __forceinline__ uint32_t rotl32(uint32_t x, int r) {
    return (x << r) | (x >> (32 - r));
}

__device__ void threefry2x32(uint32_t k0, uint32_t k1, uint32_t c0, uint32_t c1,
                             uint32_t& o0, uint32_t& o1) {
    uint32_t ks0 = k0, ks1 = k1, ks2 = k0 ^ k1 ^ 0x1BD11BDAu;
    uint32_t x0 = c0 + ks0, x1 = c1 + ks1;
    const int RA[4] = {13, 15, 26, 6};
    const int RB[4] = {17, 29, 16, 24};
#pragma unroll
    for (int i = 0; i < 4; ++i) { x0 += x1; x1 = rotl32(x1, RA[i]); x1 ^= x0; }
    x0 += ks1; x1 += ks2 + 1u;
#pragma unroll
    for (int i = 0; i < 4; ++i) { x0 += x1; x1 = rotl32(x1, RB[i]); x1 ^= x0; }
    x0 += ks2; x1 += ks0 + 2u;
#pragma unroll
    for (int i = 0; i < 4; ++i) { x0 += x1; x1 = rotl32(x1, RA[i]); x1 ^= x0; }
    x0 += ks0; x1 += ks1 + 3u;
#pragma unroll
    for (int i = 0; i < 4; ++i) { x0 += x1; x1 = rotl32(x1, RB[i]); x1 ^= x0; }
    x0 += ks1; x1 += ks2 + 4u;
#pragma unroll
    for (int i = 0; i < 4; ++i) { x0 += x1; x1 = rotl32(x1, RA[i]); x1 ^= x0; }
    x0 += ks2; x1 += ks0 + 5u;
    o0 = x0; o1 = x1;
}

// keys from jax.random.split(jax.random.key(42), 3); key(42) == (0, 42)
__device__ void jax_subkeys(int idx, uint32_t& k0, uint32_t& k1) {
    uint32_t o00, o01, o10, o11, o20, o21;
    threefry2x32(0u, 42u, 0u, 3u, o00, o01);
    threefry2x32(0u, 42u, 1u, 4u, o10, o11);
    threefry2x32(0u, 42u, 2u, 5u, o20, o21);
    // flat bits = [o00,o10,o20,o01,o11,o21], reshaped (3,2)
    uint32_t flat[6] = {o00, o10, o20, o01, o11, o21};
    k0 = flat[2 * idx];
    k1 = flat[2 * idx + 1];
}

// flat random bits element j of an n-element draw (JAX splits counts in halves)
__device__ uint32_t jax_bits(uint32_t k0, uint32_t k1, uint32_t j, uint32_t half) {
    uint32_t o0, o1;
    if (j < half) { threefry2x32(k0, k1, j, j + half, o0, o1); return o0; }
    threefry2x32(k0, k1, j - half, j, o0, o1); return o1;
}

__device__ __forceinline__ float bits_to_unit(uint32_t b) {      // [0,1)
    return __uint_as_float((b >> 9) | 0x3f800000u) - 1.0f;
}

__device__ float erfinv_f(float x) {   // Giles 2010 single-precision approximation
    float w = -__logf((1.0f - x) * (1.0f + x));
    float p;
    if (w < 5.0f) {
        w -= 2.5f;
        p = 2.81022636e-08f;
        p = fmaf(p, w, 3.43273939e-07f);
        p = fmaf(p, w, -3.5233877e-06f);
        p = fmaf(p, w, -4.39150654e-06f);
        p = fmaf(p, w, 0.00021858087f);
        p = fmaf(p, w, -0.00125372503f);
        p = fmaf(p, w, -0.00417768164f);
        p = fmaf(p, w, 0.246640727f);
        p = fmaf(p, w, 1.50140941f);
    } else {
        w = sqrtf(w) - 3.0f;
        p = -0.000200214257f;
        p = fmaf(p, w, 0.000100950558f);
        p = fmaf(p, w, 0.00134934322f);
        p = fmaf(p, w, -0.00367342844f);
        p = fmaf(p, w, 0.00573950773f);
        p = fmaf(p, w, -0.0076224613f);
        p = fmaf(p, w, 0.00943887047f);
        p = fmaf(p, w, 1.00167406f);
        p = fmaf(p, w, 2.83297682f);
    }
    return p * x;
}

// ---------------- Kernel 1: per-(b,k) parameters + peak atomics init ----------------
// Frequencies stored in Hz and phases in revolutions (the raw uniform in [0,1)),
// so the oscillator kernel can range-reduce with floorf and use the HW V_COS path.
__global__ __launch_bounds__(256) void k_params(
    const float* __restrict__ B_inh, const float* __restrict__ f0_offset,
    const float* __restrict__ A0, const float* __restrict__ tau1,
    const float* __restrict__ tau2, const float* __restrict__ a1,
    const float* __restrict__ beat_hz, const float* __restrict__ beat_depth,
    const float* __restrict__ f0, float* __restrict__ P,
    unsigned* __restrict__ peakbits, int B, int K) {
    int i = blockIdx.x * blockDim.x + threadIdx.x;
    int BK = B * K;
    if (blockIdx.x == 0 && threadIdx.x < (unsigned)B) peakbits[threadIdx.x] = 0u;
    if (i >= BK) return;
    int b = i / K, k = i % K;
    float kk = (float)(k + 1);
    float f0a = f0[b] * exp2f(f0_offset[b] * (1.0f / 1200.0f));
    float fk = f0a * kk * sqrtf(fmaf(B_inh[b], kk * kk, 1.0f));
    float fkb = fk + beat_hz[i];

    uint32_t pk0a, pk0b, pk1a, pk1b;
    jax_subkeys(0, pk0a, pk0b);
    jax_subkeys(1, pk1a, pk1b);
    uint32_t half = (uint32_t)(BK >> 1);
    float um = bits_to_unit(jax_bits(pk0a, pk0b, (uint32_t)i, half));  // phase / 2pi
    float ub = bits_to_unit(jax_bits(pk1a, pk1b, (uint32_t)i, half));

    float a = a1[i];
    P[0 * BK + i] = fk;                    // Hz
    P[1 * BK + i] = fkb;                   // Hz
    P[2 * BK + i] = um;                    // revolutions
    P[3 * BK + i] = ub;                    // revolutions
    P[4 * BK + i] = A0[i] * a;             // c1
    P[5 * BK + i] = A0[i] * (1.0f - a);    // c2
    P[6 * BK + i] = 1.0f / tau1[i];
    P[7 * BK + i] = 1.0f / tau2[i];
    P[8 * BK + i] = beat_depth[i];
}

// ---------------- Kernel 2: harmonic sum via WMMA K-reduction ----------------
// One wave = one batch b + one 16-sample time tile. A = ones(16x4) f32, B = 4x16
// panel of P[k,t]; chained V_WMMA_F32_16X16X4_F32 accumulate the K-sum in f32 with
// zero cross-lane shuffles (B/D column index == lane&15; row assignment is free
// since every A row is 1). Phase is reduced in revolutions -> HW cos fast path.
__global__ __launch_bounds__(256) void k_harmonic(
    const float* __restrict__ P, float* __restrict__ harm,
    unsigned* __restrict__ peakbits, int B, int K, int T, int totalWaves) {
    int wave = blockIdx.x * (blockDim.x >> 5) + (threadIdx.x >> 5);
    if (wave >= totalWaves) return;            // wave-uniform: EXEC stays all-ones
    int lane = threadIdx.x & 31;
    int tilesPerBatch = T >> 4;
    int b = wave / tilesPerBatch;
    int tile = wave - b * tilesPerBatch;

    int BK = B * K;
    int tcol = lane & 15;
    int tsamp = tile * 16 + tcol;
    float t = (float)tsamp * INV_SR;
    int kofs = (lane >> 4) ? 2 : 0;            // lanes 0-15 -> rows k0,k0+1; 16-31 -> k0+2,k0+3

    v2f aone; aone[0] = 1.0f; aone[1] = 1.0f;  // 16x4 all-ones A matrix
    v8f c = {0.f, 0.f, 0.f, 0.f, 0.f, 0.f, 0.f, 0.f};
    const float* Pb = P + b * K;

    for (int k0 = 0; k0 < K; k0 += 4) {
        v2f bv;
#pragma unroll
        for (int s = 0; s < 2; ++s) {
            int k = k0 + kofs + s;
            float f1 = Pb[0 * BK + k], f2 = Pb[1 * BK + k];
            float um = Pb[2 * BK + k], ub = Pb[3 * BK + k];
            float c1 = Pb[4 * BK + k], c2 = Pb[5 * BK + k];
            float r1 = Pb[6 * BK + k], r2 = Pb[7 * BK + k];
            float bd = Pb[8 * BK + k];
            float env = fmaf(c1, __expf(-t * r1), c2 * __expf(-t * r2));
            float p1 = fmaf(f1, t, um); p1 -= floorf(p1);   // phase in [0,1) rev
            float p2 = fmaf(f2, t, ub); p2 -= floorf(p2);
            float osc = __cosf(TWO_PI * p1) + bd * __cosf(TWO_PI * p2);
            bv[s] = env * osc;
        }
        // D = A(ones) x B + C : accumulates sum over 4 K-rows into every row of C
        c = __builtin_amdgcn_wmma_f32_16x16x4_f32(
            false, aone, false, bv, (short)0, c, false, false);
    }
    float h = c[0];                            // column sum for N = lane&15 (all rows equal)
    if (lane < 16) harm[b * T + tsamp] = h;

    float m = fabsf(h);
#pragma unroll
    for (int off = 16; off >= 1; off >>= 1) m = fmaxf(m, __shfl_xor(m, off, 32));
    if (lane == 0) atomicMax(peakbits + b, __float_as_uint(m));  // non-neg floats order as uints
}

// ---------------- Kernel 3: Gaussian white noise (JAX normal semantics) ----------------
__global__ __launch_bounds__(256) void k_white(float* __restrict__ white, int BT) {
    int half = BT >> 1;
    int i = blockIdx.x * blockDim.x + threadIdx.x;
    if (i >= half) return;
    uint32_t k0, k1;
    jax_subkeys(2, k0, k1);
    uint32_t o0, o1;
    threefry2x32(k0, k1, (uint32_t)i, (uint32_t)(i + half), o0, o1);
    const float lo = -0.99999994f;             // nextafter(-1, 0)
    float u0 = fmaxf(fmaf(bits_to_unit(o0), 1.0f - lo, lo), lo);
    float u1 = fmaxf(fmaf(bits_to_unit(o1), 1.0f - lo, lo), lo);
    white[i]        = 1.41421356237f * erfinv_f(u0);
    white[i + half] = 1.41421356237f * erfinv_f(u1);
}

// ---------------- Kernel 4: windowed IIR + mix + normalize + stereo ----------------
// ln(pole) = -2*pi*cutoff exactly, so W = 5/cutoff taps gives pole^W = e^-31.4 ~ 2e-14:
// the associative scan truncates to a short window (<= 501 taps, often far fewer).
// White-noise window is staged into LDS via CDNA5 async-to-LDS loads (ASYNCcnt).
__global__ __launch_bounds__(256) void k_final(
    const float* __restrict__ harm, const float* __restrict__ white,
    const float* __restrict__ noise, const float* __restrict__ width_factor,
    const unsigned* __restrict__ peakbits, float* __restrict__ out,
    int B, int T, int n, int bpb) {
    __shared__ float sh[WIN + 256];
    int b = blockIdx.x / bpb;
    int t0 = (blockIdx.x - b * bpb) << 8;
    int tid = threadIdx.x;

    float attack = fmaxf(noise[b * 4 + 0], 0.002f);
    float floor_rms = noise[b * 4 + 1];
    float centroid = noise[b * 4 + 2];
    const float f_lo = 27.5f, nyq = 24000.0f;
    float cutoff = f_lo * exp2f(centroid * log2f(nyq / f_lo)) / nyq;
    cutoff = fminf(fmaxf(cutoff, 0.01f), 0.99f);
    float pole = expf(-TWO_PI * cutoff);
    float alpha = 1.0f - pole;
    int Wn = (int)(5.0f / cutoff) + 1;         // pole^Wn <= e^-31.4
    if (Wn > WIN) Wn = WIN;
    float peak = fmaxf(__uint_as_float(peakbits[b]), 1e-6f);
    float hscale = 0.9f / peak;
    float w = fminf(fmaxf(width_factor[b], 0.0f), 2.0f);

    // Stage white[t0-WIN+1 .. t0+255] into LDS: async global->LDS for in-range
    // elements, plain zero store for the (rare) out-of-range boundary elements.
    const float* wb = white + b * T;
    for (int i = tid; i < WIN + 256; i += 256) {
        int g = t0 - (WIN - 1) + i;
        if (g >= 0 && g < T) {
            uint32_t lds_addr = (uint32_t)(uintptr_t)(lds_f32*)&sh[i];
            unsigned long long gaddr = (unsigned long long)(uintptr_t)(wb + g);
            asm volatile("global_load_async_to_lds_b32 %0, %1, off"
                         :: "v"(lds_addr), "v"(gaddr) : "memory");
        } else {
            sh[i] = 0.0f;
        }
    }
    asm volatile("s_wait_asynccnt 0x0" ::: "memory");
    __syncthreads();

    int t = t0 + tid;
    if (t < n) {
        float s = 0.0f, p = alpha;
#pragma unroll 8
        for (int j = 0; j < Wn; ++j) {         // filt[t] = sum_j pole^j * alpha * white[t-j]
            s = fmaf(p, sh[tid + (WIN - 1) - j], s);
            p *= pole;
        }
        float ts = (float)t * INV_SR;
        float nenv = __expf(-ts / attack) * floor_rms;
        float mono = fmaf(harm[b * T + t], hscale, s * nenv);
        float side = mono * w * 0.3f;
        float L = fminf(fmaxf(mono + side, -1.0f), 1.0f);
        float R = fminf(fmaxf(mono - side, -1.0f), 1.0f);
        out[(b * 2 + 0) * n + t] = L;
        out[(b * 2 + 1) * n + t] = R;
    }
}

extern "C" void kernel_launch(void* const* d_in, const int* in_sizes, int n_in,
                              void* d_out, int out_size, void* d_ws, size_t ws_size,
                              hipStream_t stream) {
    const float* B_inh      = (const float*)d_in[0];
    const float* f0_offset  = (const float*)d_in[1];
    const float* A0         = (const float*)d_in[2];
    const float* tau1       = (const float*)d_in[3];
    const float* tau2       = (const float*)d_in[4];
    const float* a1         = (const float*)d_in[5];
    const float* beat_hz    = (const float*)d_in[6];
    const float* beat_depth = (const float*)d_in[7];
    const float* noise      = (const float*)d_in[8];
    const float* f0         = (const float*)d_in[9];
    const float* width      = (const float*)d_in[10];

    int B = in_sizes[0];
    int K = in_sizes[2] / B;                    // A0 is (B,K)
    int n = out_size / (2 * B);                 // output is (B,2,n)
    int T = ((n + 239) / 240) * 240;            // multiple of 240 => multiple of 16
    int BK = B * K;
    int BT = B * T;

    // workspace layout
    float* P = (float*)d_ws;                                   // 9*BK floats
    unsigned* peakbits = (unsigned*)(P + (size_t)9 * BK);      // B uints
    float* harm = (float*)(peakbits + ((B + 3) & ~3));         // BT floats
    float* wht  = harm + (size_t)BT;                           // BT floats

    k_params<<<(BK + 255) / 256, 256, 0, stream>>>(
        B_inh, f0_offset, A0, tau1, tau2, a1, beat_hz, beat_depth, f0,
        P, peakbits, B, K);

    int totalWaves = B * (T >> 4);
    k_harmonic<<<(totalWaves + 7) / 8, 256, 0, stream>>>(
        P, harm, peakbits, B, K, T, totalWaves);

    int half = BT >> 1;
    k_white<<<(half + 255) / 256, 256, 0, stream>>>(wht, BT);

    int bpb = (n + 255) / 256;
    k_final<<<B * bpb, 256, 0, stream>>>(
        harm, wht, noise, width, peakbits, (float*)d_out, B, T, n, bpb);
}